// GAT_reg_class_51994874085708
// MI455X (gfx1250) — compile-verified
//
#include <hip/hip_runtime.h>
#include <hip/hip_bf16.h>
#include <math.h>

#define N_NODES 6144
#define N_EDGES 98304
#define IN_DIM 128
#define NODE_DIM 128
#define HIDDEN 64
#define HEADS 4
#define OUT_DIM 7
#define FEAT_DIM (HEADS * HIDDEN) /* 256 */

typedef __attribute__((ext_vector_type(2))) float v2f;
typedef __attribute__((ext_vector_type(8))) float v8f;

// ---------------------------------------------------------------- CSR build
__global__ void k_zero_i32(int* __restrict__ p, int n) {
  int i = blockIdx.x * blockDim.x + threadIdx.x;
  if (i < n) p[i] = 0;
}

__global__ void k_count_edges(const long long* __restrict__ ei, int* __restrict__ cnt, int e) {
  int i = blockIdx.x * blockDim.x + threadIdx.x;
  if (i < e) {
    int s = (int)ei[i];  // edge_index[0][i]
    atomicAdd(&cnt[s], 1);
  }
}

// single-block exclusive scan of 'cnt' (n multiple of 1024) -> row_ptr[n+1], cursor copy
__global__ void k_scan(const int* __restrict__ cnt, int* __restrict__ rowp,
                       int* __restrict__ cur, int n) {
  __shared__ int tmp[1024];
  __shared__ int carry;
  const int tid = threadIdx.x;
  if (tid == 0) carry = 0;
  __syncthreads();
  for (int base = 0; base < n; base += 1024) {
    int i = base + tid;
    int v = (i < n) ? cnt[i] : 0;
    tmp[tid] = v;
    __syncthreads();
    for (int off = 1; off < 1024; off <<= 1) {
      int t = (tid >= off) ? tmp[tid - off] : 0;
      __syncthreads();
      tmp[tid] += t;
      __syncthreads();
    }
    int incl = tmp[tid] + carry;  // inclusive prefix including old carry
    if (i < n) {
      rowp[i] = incl - v;
      cur[i] = incl - v;
    }
    __syncthreads();
    if (tid == 1023) carry = incl;
    __syncthreads();
  }
  if (tid == 0) rowp[n] = carry;
}

__global__ void k_fill_csr(const long long* __restrict__ ei, int* __restrict__ cur,
                           int* __restrict__ colx, int e) {
  int i = blockIdx.x * blockDim.x + threadIdx.x;
  if (i < e) {
    int s = (int)ei[i];
    int d = (int)ei[N_EDGES + i];
    int p = atomicAdd(&cur[s], 1);
    colx[p] = d;
  }
}

// ---------------------------------------------------------------- WMMA fp32 GEMM
// C[M,Nout] = act(A[M,K] * Wt[Nout,K]^T + bias), fp32 via V_WMMA_F32_16X16X4_F32.
// One wave per 16x16 output tile; 4 waves / block.
// A-matrix VGPR layout (16x4 f32): lane L holds M=(L&15), K = 2*(L>>4) + {0,1}.
// B-matrix VGPR layout (4x16 f32): lane L holds N=(L&15), K = 2*(L>>4) + {0,1}.
// D layout (16x16 f32, 8 VGPRs): c[r] -> row (r + 8*(L>>4)), col (L&15).
//
// Out-of-range output columns (only the Nout=7 GEMM) CLAMP their W-row pointer
// instead of branching: B lane (L&15) only feeds output column (L&15), and
// columns >= Nout are never stored, so clamped loads are harmless and the hot
// loop stays branch-free (no exec-mask save/restore per load).
__global__ void k_gemm_wt(const float* __restrict__ A, const float* __restrict__ Wt,
                          const float* __restrict__ bias, float* __restrict__ C,
                          int M, int K, int Nout, int act) {
  const int lane = threadIdx.x & 31;
  const int wave = threadIdx.x >> 5;
  const int tiles_x = (Nout + 15) >> 4;
  const int tile = blockIdx.x * (blockDim.x >> 5) + wave;
  const int tile_m = tile / tiles_x;
  const int tile_n = tile - tile_m * tiles_x;
  if (tile_m * 16 >= M) return;

  const int row0 = tile_m * 16;
  const int col0 = tile_n * 16;
  const int mrow = row0 + (lane & 15);
  int wrow = col0 + (lane & 15);  // Wt row == output column
  if (wrow > Nout - 1) wrow = Nout - 1;  // clamp: keeps loads in-bounds, no divergence
  const int koff = (lane >> 4) * 2;

  const float* pa = A + (size_t)mrow * K + koff;
  const float* pb = Wt + (size_t)wrow * K + koff;

  v8f c = {};
#pragma unroll 8
  for (int k0 = 0; k0 < K; k0 += 4) {
    __builtin_prefetch(pa + k0 + 32, 0, 3);  // global_prefetch_b8
    v2f a = *(const v2f*)(pa + k0);
    v2f b = *(const v2f*)(pb + k0);
    // D = A*B + C   (fp32 WMMA, K=4 step)
    c = __builtin_amdgcn_wmma_f32_16x16x4_f32(false, a, false, b, (short)0, c, false, false);
  }

  const int scol = col0 + (lane & 15);
  if (scol < Nout) {
    float bv = bias ? bias[scol] : 0.0f;
    const int rbase = row0 + ((lane >> 4) << 3);
#pragma unroll
    for (int r = 0; r < 8; ++r) {
      float v = c[r] + bv;
      if (act == 1) v = fmaxf(v, 0.0f);
      C[(size_t)(rbase + r) * Nout + scol] = v;
    }
  }
}

// ---------------------------------------------------------------- f1/f2 projections
__global__ void k_f1f2(const float* __restrict__ XH, const float* __restrict__ a1,
                       const float* __restrict__ a2, float* __restrict__ F1,
                       float* __restrict__ F2) {
  int t = blockIdx.x * blockDim.x + threadIdx.x;
  if (t >= N_NODES * HEADS) return;
  int n = t >> 2;
  int h = t & 3;
  const float* xr = XH + (size_t)n * FEAT_DIM + h * HIDDEN;
  const float* w1 = a1 + h * HIDDEN;
  const float* w2 = a2 + h * HIDDEN;
  float s1 = 0.f, s2 = 0.f;
#pragma unroll 8
  for (int o = 0; o < HIDDEN; ++o) {
    float x = xr[o];
    s1 += x * w1[o];
    s2 += x * w2[o];
  }
  F1[h * N_NODES + n] = s1;
  F2[h * N_NODES + n] = s2;
}

// ---------------------------------------------------------------- sparse GAT attention
// One wave per (node, head). Online (flash) softmax over unique neighbors with
// duplicate-edge multiplicity folded into the logit (matches dense ref exactly:
// e = leaky_relu(f1+f2) * adj_count; non-neighbors exp-underflow to 0).
__global__ void k_attn(const int* __restrict__ rowp, const int* __restrict__ colx,
                       const float* __restrict__ F1, const float* __restrict__ F2,
                       const float* __restrict__ XH, float* __restrict__ FEAT) {
  const int n = blockIdx.x;
  const int hh = threadIdx.x >> 5;
  const int lane = threadIdx.x & 31;

  const int start = rowp[n];
  const int end = rowp[n + 1];
  const float f1n = F1[hh * N_NODES + n];

  float m = -INFINITY, s = 0.0f, acc0 = 0.0f, acc1 = 0.0f;

  auto process = [&](int j, float cntf) {
    float l = f1n + F2[hh * N_NODES + j];
    l = (l > 0.0f) ? l : 0.01f * l;  // leaky_relu(0.01)
    l *= cntf;                       // duplicate edges scale the logit
    float mnew = fmaxf(m, l);
    float scale = __expf(m - mnew);  // exp(-inf)=0 on first entry
    float w = __expf(l - mnew);
    const float* xr = XH + (size_t)j * FEAT_DIM + hh * HIDDEN;
    s = s * scale + w;
    acc0 = acc0 * scale + w * xr[lane];
    acc1 = acc1 * scale + w * xr[lane + 32];
    m = mnew;
  };

  // self-loop: count = 1 (from +eye) + duplicates of (n,n) in edge list
  int selfdup = 0;
  for (int k = start; k < end; ++k)
    if (colx[k] == n) ++selfdup;
  process(n, 1.0f + (float)selfdup);

  for (int k = start; k < end; ++k) {
    int j = colx[k];
    if (j == n) continue;
    bool first = true;
    int cnt = 1;
    for (int k2 = start; k2 < end; ++k2) {
      if (colx[k2] == j) {
        if (k2 < k) { first = false; break; }
        if (k2 > k) ++cnt;
      }
    }
    if (!first) continue;
    process(j, (float)cnt);
  }

  float o0 = acc0 / s;
  float o1 = acc1 / s;
  o0 = (o0 > 0.0f) ? o0 : (__expf(o0) - 1.0f);  // elu
  o1 = (o1 > 0.0f) ? o1 : (__expf(o1) - 1.0f);
  float* fr = FEAT + (size_t)n * FEAT_DIM + hh * HIDDEN;
  fr[lane] = o0;
  fr[lane + 32] = o1;
}

// ---------------------------------------------------------------- host launcher
static inline int gemm_blocks(int M, int Nout) {
  int tiles = (M / 16) * ((Nout + 15) / 16);
  return (tiles + 3) / 4;  // 4 waves (tiles) per 128-thread block
}

extern "C" void kernel_launch(void* const* d_in, const int* in_sizes, int n_in,
                              void* d_out, int out_size, void* d_ws, size_t ws_size,
                              hipStream_t stream) {
  const float* x      = (const float*)d_in[0];
  const long long* ei = (const long long*)d_in[1];  // int64 edge_index [2,E]
  const float* U0_W   = (const float*)d_in[2];      // [128,128]
  const float* W      = (const float*)d_in[3];      // [4,64,128] == [256,128]
  const float* a1     = (const float*)d_in[4];      // [4,64]
  const float* a2     = (const float*)d_in[5];
  const float* fc0_W  = (const float*)d_in[6];      // [64,256]
  const float* fc0_b  = (const float*)d_in[7];
  const float* fc1_W  = (const float*)d_in[8];      // [64,64]
  const float* fc1_b  = (const float*)d_in[9];
  const float* fc2_W  = (const float*)d_in[10];     // [7,64]
  const float* fc2_b  = (const float*)d_in[11];
  float* out = (float*)d_out;

  char* ws = (char*)d_ws;
  auto alloc = [&](size_t bytes) -> void* {
    void* p = (void*)ws;
    ws += (bytes + 255) & ~(size_t)255;
    return p;
  };
  float* h    = (float*)alloc((size_t)N_NODES * NODE_DIM * 4);
  float* xh   = (float*)alloc((size_t)N_NODES * FEAT_DIM * 4);
  float* feat = (float*)alloc((size_t)N_NODES * FEAT_DIM * 4);
  float* F1   = (float*)alloc((size_t)HEADS * N_NODES * 4);
  float* F2   = (float*)alloc((size_t)HEADS * N_NODES * 4);
  float* y0   = (float*)alloc((size_t)N_NODES * HIDDEN * 4);
  float* y1   = (float*)alloc((size_t)N_NODES * HIDDEN * 4);
  int* cnt    = (int*)alloc((size_t)N_NODES * 4);
  int* cur    = (int*)alloc((size_t)N_NODES * 4);
  int* rowp   = (int*)alloc((size_t)(N_NODES + 1) * 4);
  int* colx   = (int*)alloc((size_t)N_EDGES * 4);

  // --- CSR build ---
  k_zero_i32<<<(N_NODES + 255) / 256, 256, 0, stream>>>(cnt, N_NODES);
  k_count_edges<<<(N_EDGES + 255) / 256, 256, 0, stream>>>(ei, cnt, N_EDGES);
  k_scan<<<1, 1024, 0, stream>>>(cnt, rowp, cur, N_NODES);
  k_fill_csr<<<(N_EDGES + 255) / 256, 256, 0, stream>>>(ei, cur, colx, N_EDGES);

  // --- h = x @ U0_W^T  [6144,128] ---
  k_gemm_wt<<<gemm_blocks(N_NODES, NODE_DIM), 128, 0, stream>>>(
      x, U0_W, nullptr, h, N_NODES, IN_DIM, NODE_DIM, 0);
  // --- xh = h @ Wcat^T [6144,256] (col = head*64 + o) ---
  k_gemm_wt<<<gemm_blocks(N_NODES, FEAT_DIM), 128, 0, stream>>>(
      h, W, nullptr, xh, N_NODES, NODE_DIM, FEAT_DIM, 0);
  // --- f1, f2 ---
  k_f1f2<<<(N_NODES * HEADS + 255) / 256, 256, 0, stream>>>(xh, a1, a2, F1, F2);
  // --- sparse attention + ELU -> feat [6144,256] ---
  k_attn<<<N_NODES, HEADS * 32, 0, stream>>>(rowp, colx, F1, F2, xh, feat);
  // --- MLP head ---
  k_gemm_wt<<<gemm_blocks(N_NODES, HIDDEN), 128, 0, stream>>>(
      feat, fc0_W, fc0_b, y0, N_NODES, FEAT_DIM, HIDDEN, 1);
  k_gemm_wt<<<gemm_blocks(N_NODES, HIDDEN), 128, 0, stream>>>(
      y0, fc1_W, fc1_b, y1, N_NODES, HIDDEN, HIDDEN, 1);
  k_gemm_wt<<<gemm_blocks(N_NODES, OUT_DIM), 128, 0, stream>>>(
      y1, fc2_W, fc2_b, out, N_NODES, HIDDEN, OUT_DIM, 0);
}